// Helmholtz_75539884802248
// MI455X (gfx1250) — compile-verified
//
#include <hip/hip_runtime.h>

// Helmholtz operator: out = lap(x) - kappa^2 * OMEGA * (OMEGA - i*gamma) * x
// x complex64 (float2), kappa/gamma f32, shape (B,1,1024,1024), h = 1/1024.
//
// Bandwidth-bound (~201 MB/call -> ~8.6us floor @ 23.3 TB/s).
// CDNA5 path: async global->LDS staging (global_load_async_to_lds_b64 +
// s_wait_asynccnt) of the x tile + halo; conflict-free ds_load_b64 stencil
// reads; coalesced b32/b64 global traffic for kappa/gamma/out.

#ifndef OMEGA_F
#define OMEGA_F 80.0f
#endif

constexpr int M_DIM = 1024;
constexpr int N_DIM = 1024;
constexpr int TX = 64;          // tile width (columns)
constexpr int TY = 16;          // tile height (rows)
constexpr int LW = TX + 2;      // 66 (halo)
constexpr int LH = TY + 2;      // 18 (halo)
constexpr int NTHREADS = 256;   // 8 wave32
constexpr int NSTAGE = (LH * LW + NTHREADS - 1) / NTHREADS;  // 5

__device__ __forceinline__ unsigned lds_offset_of(const void* p) {
    // Flat shared-aperture address: low 32 bits are the LDS byte offset
    return (unsigned)(size_t)p;
}

__global__ __launch_bounds__(NTHREADS) void helmholtz_async_stencil(
    const float2* __restrict__ x,
    const float*  __restrict__ kappa,
    const float*  __restrict__ gamma_,
    float2* __restrict__ out)
{
    __shared__ float2 tile[LH * LW];   // 9504 B

    const int tid  = threadIdx.x;
    const int col0 = blockIdx.x * TX;
    const int row0 = blockIdx.y * TY;
    const size_t imgBase = (size_t)blockIdx.z * ((size_t)M_DIM * N_DIM);

    // ---- Stage x tile (+halo) into LDS with CDNA5 async copies ----
    // 18*66 = 1188 entries; fixed 5-deep unrolled staging, <=5 per thread.
#pragma unroll
    for (int i = 0; i < NSTAGE; ++i) {
        const int e = tid + i * NTHREADS;
        if (e < LH * LW) {
            const int ly = e / LW;               // const-div -> mul/shift
            const int lx = e - ly * LW;
            const int gr = row0 + ly - 1;
            const int gc = col0 + lx - 1;
            const unsigned lds_off = lds_offset_of(&tile[e]);
            if ((unsigned)gr < (unsigned)M_DIM && (unsigned)gc < (unsigned)N_DIM) {
                const float2* gp = x + imgBase + (size_t)gr * N_DIM + gc;
                // vdst = LDS byte offset, vaddr = 64-bit global address
                asm volatile("global_load_async_to_lds_b64 %0, %1, off"
                             :
                             : "v"(lds_off), "v"(gp)
                             : "memory");
            } else {
                tile[e] = make_float2(0.0f, 0.0f);  // zero padding at border
            }
        }
    }
    // Wait for this wave's async LDS writes, then workgroup barrier
    // (barrier covers the DS zero-fill stores via DScnt).
    asm volatile("s_wait_asynccnt 0x0" ::: "memory");
    __syncthreads();

    // ---- Compute 4 consecutive rows per thread with vertical reuse ----
    const float inv_h2 = (float)M_DIM * (float)M_DIM;  // 1/h^2 = m^2
    const int lx  = tid & (TX - 1);      // 0..63 column within tile
    const int tg  = tid >> 6;            // 0..3  row group
    const int ly0 = tg * 4;              // first computed row in tile coords

    // Rolling 3-row window of the center column (tile col lx+1).
    float2 rm1 = tile[(ly0    ) * LW + (lx + 1)];   // row above first point
    float2 rc  = tile[(ly0 + 1) * LW + (lx + 1)];   // first point

#pragma unroll
    for (int k = 0; k < 4; ++k) {
        const int ly = ly0 + k;
        const float2 rp1 = tile[(ly + 2) * LW + (lx + 1)];  // row below
        const float2 lf  = tile[(ly + 1) * LW + (lx    )];
        const float2 rt  = tile[(ly + 1) * LW + (lx + 2)];

        const size_t gi = imgBase + (size_t)(row0 + ly) * N_DIM + (col0 + lx);
        const float kap = kappa[gi];
        const float gam = gamma_[gi];

        const float lap_re = (4.0f * rc.x - rm1.x - rp1.x - lf.x - rt.x) * inv_h2;
        const float lap_im = (4.0f * rc.y - rm1.y - rp1.y - lf.y - rt.y) * inv_h2;

        // helm = k^2*w^2 - i*k^2*w*g ; out = lap - helm*x
        const float k2w = kap * kap * OMEGA_F;   // k^2 * w
        const float A   = k2w * OMEGA_F;         // k^2 * w^2
        const float Bv  = k2w * gam;             // k^2 * w * g

        float2 o;
        o.x = lap_re - A * rc.x - Bv * rc.y;
        o.y = lap_im - A * rc.y + Bv * rc.x;
        out[gi] = o;

        rm1 = rc;         // slide the vertical window
        rc  = rp1;
    }
}

extern "C" void kernel_launch(void* const* d_in, const int* in_sizes, int n_in,
                              void* d_out, int out_size, void* d_ws, size_t ws_size,
                              hipStream_t stream) {
    const float2* x      = (const float2*)d_in[0];   // complex64 -> float2
    const float*  kappa  = (const float*)d_in[1];
    const float*  gamma_ = (const float*)d_in[2];
    float2*       out    = (float2*)d_out;

    // Batch count from kappa's flat element count (robust to B changes).
    const int B = in_sizes[1] / (M_DIM * N_DIM);

    dim3 grid(N_DIM / TX, M_DIM / TY, B);
    helmholtz_async_stencil<<<grid, NTHREADS, 0, stream>>>(x, kappa, gamma_, out);
}